// ResMRConv_59150289600865
// MI455X (gfx1250) — compile-verified
//
#include <hip/hip_runtime.h>

#define N_NODES 50000
#define N_EDGES 640000
#define WIDTH   128
#define KDIM    256   // 2*WIDTH

typedef __attribute__((ext_vector_type(2))) float v2f;
typedef __attribute__((ext_vector_type(8))) float v8f;

// ---- monotone f32 <-> u32 mapping so we can use native GLOBAL_ATOMIC_MAX_U32
__device__ __forceinline__ unsigned enc_f32(float f) {
    int i = __float_as_int(f);
    return (i >= 0) ? ((unsigned)i | 0x80000000u) : ~(unsigned)i;
}
__device__ __forceinline__ float dec_f32(unsigned u) {
    if (u == 0u) return 0.0f;                       // empty-segment sentinel -> 0
    return (u & 0x80000000u) ? __uint_as_float(u & 0x7fffffffu)
                             : __uint_as_float(~u);
}

// ---- kernel 1: init encoded-max buffer to sentinel 0 (== -inf in encoding)
__global__ void init_maxes(unsigned* __restrict__ m, int n) {
    int i = blockIdx.x * blockDim.x + threadIdx.x;
    int stride = gridDim.x * blockDim.x;
    for (; i < n; i += stride) m[i] = 0u;
}

// ---- kernel 2: one wave32 per edge; lane handles 4 consecutive features.
// diffs = x[dst] - x[src]; scatter-max into maxes[dst] via u32 atomic max.
__global__ __launch_bounds__(256) void scatter_max_kernel(
        const float* __restrict__ x, const int* __restrict__ e,
        unsigned* __restrict__ mx) {
    int edge = (blockIdx.x * 256 + threadIdx.x) >> 5;   // global wave id
    int lane = threadIdx.x & 31;
    if (edge >= N_EDGES) return;
    int s = e[edge];
    int d = e[N_EDGES + edge];
    float4 a = ((const float4*)(x + (size_t)d * WIDTH))[lane];
    float4 b = ((const float4*)(x + (size_t)s * WIDTH))[lane];
    unsigned* mp = mx + (size_t)d * WIDTH + lane * 4;
    atomicMax(&mp[0], enc_f32(a.x - b.x));
    atomicMax(&mp[1], enc_f32(a.y - b.y));
    atomicMax(&mp[2], enc_f32(a.z - b.z));
    atomicMax(&mp[3], enc_f32(a.w - b.w));
}

// ---- kernel 3: out = x + relu([x || maxes] @ W^T + b), f32 WMMA 16x16x4.
// Block = 256 threads = 8 waves; block owns 16 rows, each wave a 16-col slab.
__global__ __launch_bounds__(256) void gemm_relu_res_kernel(
        const float* __restrict__ x, const unsigned* __restrict__ mx,
        const float* __restrict__ W, const float* __restrict__ bias,
        float* __restrict__ out) {
    constexpr int LDH = KDIM + 2;          // pad -> conflict-free ds_load_b64
    __shared__ float h[16 * LDH];

    const int r0  = blockIdx.x * 16;
    const int tid = threadIdx.x;

    // Stage h tile: thread loads 16 floats of row tid>>4, col-group tid&15.
    {
        int row = tid >> 4;
        int cg  = tid & 15;
        int col = cg * 16;
        size_t node = (size_t)(r0 + row);
        float* dst = &h[row * LDH + col];
        if (cg < 8) {                                   // x half
            const float4* src = (const float4*)(x + node * WIDTH + col);
            #pragma unroll
            for (int j = 0; j < 4; ++j) {
                float4 v = src[j];
                dst[4*j+0] = v.x; dst[4*j+1] = v.y;
                dst[4*j+2] = v.z; dst[4*j+3] = v.w;
            }
        } else {                                        // decoded maxes half
            const uint4* src = (const uint4*)(mx + node * WIDTH + (col - WIDTH));
            #pragma unroll
            for (int j = 0; j < 4; ++j) {
                uint4 v = src[j];
                dst[4*j+0] = dec_f32(v.x); dst[4*j+1] = dec_f32(v.y);
                dst[4*j+2] = dec_f32(v.z); dst[4*j+3] = dec_f32(v.w);
            }
        }
    }
    __syncthreads();

    const int lane  = tid & 31;
    const int c0    = (tid >> 5) * 16;        // wave's output-column base
    const int n     = c0 + (lane & 15);       // output feature (= W row)
    const int khalf = (lane >> 4) * 2;        // lanes 16-31 hold K+2,K+3
    const int mrow  = lane & 15;              // A-fragment row
    const float* wrow = W + (size_t)n * KDIM;

    v8f acc = {};
    #pragma unroll 8
    for (int k0 = 0; k0 < KDIM; k0 += 4) {
        int kk = k0 + khalf;
        v2f a;                                 // A: h[mrow][kk..kk+1] from LDS
        a.x = h[mrow * LDH + kk];
        a.y = h[mrow * LDH + kk + 1];
        v2f bf = *(const v2f*)(wrow + kk);     // B: W[n][kk..kk+1]
        acc = __builtin_amdgcn_wmma_f32_16x16x4_f32(
                  false, a, false, bf, (short)0, acc, false, false);
    }

    const float bn    = bias[n];
    const int   mbase = (lane >> 4) * 8;       // D: VGPR v -> row v + mbase
    #pragma unroll
    for (int v = 0; v < 8; ++v) {
        int m = mbase + v;
        float val = acc[v] + bn;
        val = val > 0.0f ? val : 0.0f;
        out[(size_t)(r0 + m) * WIDTH + n] = h[m * LDH + n] + val;  // +x residual
    }
}

extern "C" void kernel_launch(void* const* d_in, const int* in_sizes, int n_in,
                              void* d_out, int out_size, void* d_ws, size_t ws_size,
                              hipStream_t stream) {
    const float* x  = (const float*)d_in[0];
    const int*   e  = (const int*)d_in[1];
    const float* W  = (const float*)d_in[2];
    const float* b  = (const float*)d_in[3];
    float*    out   = (float*)d_out;
    unsigned* mx    = (unsigned*)d_ws;        // N_NODES*WIDTH encoded maxes

    init_maxes<<<1024, 256, 0, stream>>>(mx, N_NODES * WIDTH);
    scatter_max_kernel<<<N_EDGES / 8, 256, 0, stream>>>(x, e, mx);
    gemm_relu_res_kernel<<<N_NODES / 16, 256, 0, stream>>>(x, mx, W, b, out);
}